// HidePatchLayer_11493332484436
// MI455X (gfx1250) — compile-verified
//
#include <hip/hip_runtime.h>

// Hide-and-Seek patch masking for MI455X (gfx1250).
//
// out[b,w,h,c] = in[b,w,h,c] * (cell_rand[w/16, h/16] > 0.5 ? 1 : 0)
//
// Roofline: 38.5M f32 elements -> 308 MB HBM traffic for ~38.5 MFLOP
// (AI = 0.125 FLOP/byte). Pure streaming, floor ~13.2 us at 23.3 TB/s.
// Strategy:
//  - 128-bit non-temporal loads/stores (dataset > 192 MB L2, zero reuse)
//  - 4 block-strided float4 per thread -> clause of 4 NT b128 loads in
//    flight per wave before the first s_wait_loadcnt (MLP at fixed occupancy)
//  - 14x14 keep-mask staged once per block in LDS
//  - no WMMA: elementwise op, no contraction dim; matrix units cannot
//    reduce the byte count that bounds this kernel.

typedef float v4f __attribute__((ext_vector_type(4)));

#define HIDE_PROB 0.5f

constexpr int kB  = 256;
constexpr int kW  = 224;
constexpr int kH  = 224;
constexpr int kC  = 3;
constexpr int kGS = 16;

constexpr int kNX        = kW / kGS;            // 14 cells along W
constexpr int kNY        = kH / kGS;            // 14 cells along H
constexpr int kRowFloats = kH * kC;             // 672 contiguous floats per (b,w) row
constexpr int kRowVec4   = kRowFloats / 4;      // 168 float4 per row
constexpr int kVec4HCell = kGS * kC / 4;        // 12 float4 per h-cell (48 floats,
                                                // 48 % 4 == 0 -> no straddling)
constexpr int kTotalVec4 = kB * kW * kRowVec4;  // 9,633,792

constexpr int kBlock    = 256;                  // 8 wave32s
constexpr int kPerThr   = 4;                    // 4 x float4 = 64 B per thread
constexpr int kBlockV4  = kBlock * kPerThr;     // 1024 float4 per block
constexpr int kBlocks   = kTotalVec4 / kBlockV4; // 9,408 exact, no tail

__global__ __launch_bounds__(kBlock) void hide_patch_kernel(
    const v4f* __restrict__ in,
    const float* __restrict__ cell_rand,
    v4f* __restrict__ out) {
  // Stage the 14x14 keep table in LDS as 0.0f / 1.0f.
  __shared__ float keep[kNX * kNY];
  const int t = threadIdx.x;
  if (t < kNX * kNY) {
    keep[t] = (cell_rand[t] > HIDE_PROB) ? 1.0f : 0.0f;
  }
  __syncthreads();

  const int base = blockIdx.x * kBlockV4 + t;   // block-strided, fully coalesced

  // Gather the 4 vectors first so the compiler can clause the NT loads.
  v4f v[kPerThr];
  float m[kPerThr];
#pragma unroll
  for (int j = 0; j < kPerThr; ++j) {
    const int i = base + j * kBlock;
    v[j] = __builtin_nontemporal_load(&in[i]);

    // i -> (row = b*W + w, p4 within the 672-float row); const-divisor math.
    const int row = i / kRowVec4;
    const int p4  = i - row * kRowVec4;         // 0..167
    const int w   = row % kW;                   // 0..223
    const int cx  = w >> 4;                     // w / 16
    const int cy  = p4 / kVec4HCell;            // h / 16
    m[j] = keep[cx * kNY + cy];
  }

#pragma unroll
  for (int j = 0; j < kPerThr; ++j) {
    v[j] *= m[j];
    __builtin_nontemporal_store(v[j], &out[base + j * kBlock]);
  }
}

extern "C" void kernel_launch(void* const* d_in, const int* in_sizes, int n_in,
                              void* d_out, int out_size, void* d_ws, size_t ws_size,
                              hipStream_t stream) {
  (void)in_sizes; (void)n_in; (void)d_ws; (void)ws_size; (void)out_size;

  const v4f*   in       = (const v4f*)d_in[0];   // inputs [256,224,224,3] f32
  const float* cell_rnd = (const float*)d_in[1]; // cell_rand [14,14] f32
  // d_in[2] = grid_size scalar (16), baked in as a compile-time constant.
  v4f* out = (v4f*)d_out;

  hide_patch_kernel<<<kBlocks, kBlock, 0, stream>>>(in, cell_rnd, out);
}